// StentMultiPredictor_10342281249011
// MI455X (gfx1250) — compile-verified
//
#include <hip/hip_runtime.h>
#include <hip/hip_bf16.h>

// ---------------------------------------------------------------------------
// GraphSAGE x2 + BN + ReLU + Linear, fp32 end-to-end using V_WMMA_F32_16X16X4_F32
// Layer-2 A-tiles staged with GLOBAL_LOAD_ASYNC_TO_LDS_B128 (ASYNCcnt path).
// N=50000, E=800000, C_IN=4, H=256, C_OUT=4
// ---------------------------------------------------------------------------

#define H_DIM 256
#define EPS 1e-5f

typedef __attribute__((ext_vector_type(2))) float v2f;
typedef __attribute__((ext_vector_type(8))) float v8f;

__device__ __forceinline__ v8f wmma_f32_k4(v2f a, v2f b, v8f c) {
  // D = A(16x4 f32) * B(4x16 f32) + C(16x16 f32), full fp32 precision
  return __builtin_amdgcn_wmma_f32_16x16x4_f32(
      /*neg_a=*/false, a, /*neg_b=*/false, b,
      /*c_mod=*/(short)0, c, /*reuse_a=*/false, /*reuse_b=*/false);
}

// Async global -> LDS copy, 16 bytes per lane, tracked by ASYNCcnt.
// LDS operand: generic shared pointer truncated to 32 bits == LDS byte offset
// (ISA 10.2: flat LDS aperture carries offset in addr[31:0]).
__device__ __forceinline__ void async_load_b128(const void* g, void* lds) {
  uint64_t ga = (uint64_t)(uintptr_t)g;
  uint32_t la = (uint32_t)(uintptr_t)lds;
  asm volatile("global_load_async_to_lds_b128 %0, %1, off"
               :: "v"(la), "v"(ga) : "memory");
}

__device__ __forceinline__ void wait_asynccnt0() {
  asm volatile("s_wait_asynccnt 0" ::: "memory");
}

// ---------------------------------------------------------------------------
// 0) zero scratch (grid-stride)
// ---------------------------------------------------------------------------
__global__ void k_zero(float* __restrict__ p, long n) {
  long i = (long)blockIdx.x * blockDim.x + threadIdx.x;
  long stride = (long)gridDim.x * blockDim.x;
  for (; i < n; i += stride) p[i] = 0.0f;
}

// ---------------------------------------------------------------------------
// 1) layer-1 edge scatter: agg1[dst] += x[src]*scale ; cnt[dst] += 1
// ---------------------------------------------------------------------------
__global__ void k_scatter1(const float* __restrict__ x,
                           const int* __restrict__ ei,
                           float* __restrict__ agg1,
                           float* __restrict__ cnt,
                           long E) {
  long e = (long)blockIdx.x * blockDim.x + threadIdx.x;
  if (e >= E) return;
  int src = ei[e];
  int dst = ei[E + e];
  const float* xs = x + (size_t)src * 4;
  float* ag = agg1 + (size_t)dst * 4;
  atomicAdd(ag + 0, xs[0] * 1000.0f);
  atomicAdd(ag + 1, xs[1] * 1000.0f);
  atomicAdd(ag + 2, xs[2] * 1000.0f);
  atomicAdd(ag + 3, xs[3]);
  atomicAdd(cnt + dst, 1.0f);
}

// ---------------------------------------------------------------------------
// 2) layer-1 GEMM (K=4, single WMMA per fragment pair) + BN1 stats
//    z1 = mean1 @ W1l^T + b1l + (x*scale) @ W1r^T
//    block = 256 threads = 8 waves; block covers 16 nodes x 256 channels.
// ---------------------------------------------------------------------------
__global__ void k_layer1_wmma(const float* __restrict__ x,
                              const float* __restrict__ agg1,
                              const float* __restrict__ cnt,
                              const float* __restrict__ W1l,
                              const float* __restrict__ b1l,
                              const float* __restrict__ W1r,
                              float* __restrict__ z1,
                              float* __restrict__ sum1,
                              float* __restrict__ sq1,
                              int N) {
  const int lane = threadIdx.x & 31;
  const int wave = threadIdx.x >> 5;
  const int tile = blockIdx.x;                 // 16-node row tile
  const int m = lane & 15;
  const int kb = (lane >> 4) << 1;             // 0 or 2
  const int mr = tile * 16 + m;
  const bool full = (tile * 16 + 16) <= N;

  // A fragments: mean-aggregate and scaled self features (16x4 each)
  v2f a_mean, a_self;
  if (mr < N) {
    float cm = fmaxf(cnt[mr], 1.0f);
    a_mean.x = agg1[(size_t)mr * 4 + kb] / cm;
    a_mean.y = agg1[(size_t)mr * 4 + kb + 1] / cm;
    float s0 = (kb     < 3) ? 1000.0f : 1.0f;
    float s1 = (kb + 1 < 3) ? 1000.0f : 1.0f;
    a_self.x = x[(size_t)mr * 4 + kb] * s0;
    a_self.y = x[(size_t)mr * 4 + kb + 1] * s1;
  } else {
    a_mean.x = a_mean.y = 0.0f;
    a_self.x = a_self.y = 0.0f;
  }

  const int n = lane & 15;
#pragma unroll
  for (int sub = 0; sub < 2; ++sub) {
    const int col0 = wave * 32 + sub * 16;
    const int gcol = col0 + n;
    // B fragments: W^T tiles (4 x 16). W is [256,4] row-major.
    v2f bl, br;
    bl.x = W1l[(size_t)gcol * 4 + kb];
    bl.y = W1l[(size_t)gcol * 4 + kb + 1];
    br.x = W1r[(size_t)gcol * 4 + kb];
    br.y = W1r[(size_t)gcol * 4 + kb + 1];

    v8f acc = {};
    acc = wmma_f32_k4(a_mean, bl, acc);
    acc = wmma_f32_k4(a_self, br, acc);

    const float bias = b1l[gcol];
    float s = 0.0f, sq = 0.0f;
    const int rbase = tile * 16 + ((lane >> 4) << 3);
    if (full) {
#pragma unroll
      for (int r = 0; r < 8; ++r) {
        float v = acc[r] + bias;
        z1[(size_t)(rbase + r) * H_DIM + gcol] = v;
        s += v;
        sq += v * v;
      }
    } else {
      for (int r = 0; r < 8; ++r) {
        int row = rbase + r;
        if (row < N) {
          float v = acc[r] + bias;
          z1[(size_t)row * H_DIM + gcol] = v;
          s += v;
          sq += v * v;
        }
      }
    }
    atomicAdd(&sum1[gcol], s);
    atomicAdd(&sq1[gcol], sq);
  }
}

// ---------------------------------------------------------------------------
// 3) BN + ReLU elementwise: h = relu((z-mu)*rsqrt(var+eps)*g + b)
// ---------------------------------------------------------------------------
__global__ void k_bn_relu(const float* __restrict__ z,
                          const float* __restrict__ sum,
                          const float* __restrict__ sq,
                          const float* __restrict__ gamma,
                          const float* __restrict__ beta,
                          float* __restrict__ h,
                          long total, float invN) {
  long i = (long)blockIdx.x * blockDim.x + threadIdx.x;
  long stride = (long)gridDim.x * blockDim.x;
  for (; i < total; i += stride) {
    int j = (int)(i & (H_DIM - 1));
    float mu = sum[j] * invN;
    float var = sq[j] * invN - mu * mu;
    float v = (z[i] - mu) * rsqrtf(var + EPS) * gamma[j] + beta[j];
    h[i] = fmaxf(v, 0.0f);
  }
}

// ---------------------------------------------------------------------------
// 4) layer-2 edge scatter: agg2[dst][:] += h1[src][:]  (256 ch, float4 loads)
//    block = 256 threads = 4 edges x 64 lanes x 4 channels
// ---------------------------------------------------------------------------
__global__ void k_scatter2(const float* __restrict__ h1,
                           const int* __restrict__ ei,
                           float* __restrict__ agg2,
                           long E) {
  long e = (long)blockIdx.x * 4 + (threadIdx.x >> 6);
  if (e >= E) return;
  int li = threadIdx.x & 63;                   // 64 lanes * 4 ch = 256
  int src = ei[e];
  int dst = ei[E + e];
  const float4 v = *(const float4*)(h1 + (size_t)src * H_DIM + li * 4);
  float* ag = agg2 + (size_t)dst * H_DIM + li * 4;
  atomicAdd(ag + 0, v.x);
  atomicAdd(ag + 1, v.y);
  atomicAdd(ag + 2, v.z);
  atomicAdd(ag + 3, v.w);
}

// ---------------------------------------------------------------------------
// 5) layer-2 dual GEMM via WMMA f32 16x16x4, K=256 (64 k-steps per matrix).
//    A tiles staged into LDS with GLOBAL_LOAD_ASYNC_TO_LDS_B128 (full tiles);
//    mean division folded into fragment load as a reciprocal multiply.
//    z2 = mean2@W2l^T + b2l + h1@W2r^T, plus BN2 statistics.
// ---------------------------------------------------------------------------
#define LDS_PITCH (H_DIM + 4)   // 260 floats = 1040 B rows (16 B aligned)
__global__ void k_layer2_wmma(const float* __restrict__ agg2,
                              const float* __restrict__ cnt,
                              const float* __restrict__ h1,
                              const float* __restrict__ W2l,
                              const float* __restrict__ b2l,
                              const float* __restrict__ W2r,
                              float* __restrict__ z2,
                              float* __restrict__ sum2,
                              float* __restrict__ sq2,
                              int N) {
  __shared__ float sM[16][LDS_PITCH];          // raw aggregated features
  __shared__ float sH[16][LDS_PITCH];          // self features

  const int tile = blockIdx.x;
  const int t = threadIdx.x;
  const bool full = (tile * 16 + 16) <= N;

  if (full) {
    // 16 rows x 64 float4-chunks per array = 1024 chunks; lane handles
    // chunks t, t+256, t+512, t+768 -> 4 async b128 issues per array.
#pragma unroll
    for (int i = 0; i < 4; ++i) {
      int ch = t + i * 256;
      int row = ch >> 6;                       // 0..15
      int c4 = ch & 63;                        // float4 chunk in row
      size_t goff = (size_t)(tile * 16 + row) * H_DIM + c4 * 4;
      async_load_b128(agg2 + goff, &sM[row][c4 * 4]);
      async_load_b128(h1 + goff, &sH[row][c4 * 4]);
    }
    wait_asynccnt0();                          // this wave's DMAs done
  } else {
#pragma unroll
    for (int i = 0; i < 16; ++i) {
      int idx = t + i * 256;
      int row = idx >> 8;
      int col = idx & (H_DIM - 1);
      int gr = tile * 16 + row;
      if (gr < N) {
        sM[row][col] = agg2[(size_t)gr * H_DIM + col];
        sH[row][col] = h1[(size_t)gr * H_DIM + col];
      } else {
        sM[row][col] = 0.0f;
        sH[row][col] = 0.0f;
      }
    }
  }
  __syncthreads();

  const int lane = t & 31;
  const int wave = t >> 5;
  const int m = lane & 15;
  const int khalf = (lane >> 4) << 1;          // 0 or 2
  const int n = lane & 15;

  // per-lane reciprocal of in-degree for A-row m (folds mean into frag load)
  int mr = tile * 16 + m;
  if (mr >= N) mr = N - 1;
  const float rcpc = 1.0f / fmaxf(cnt[mr], 1.0f);

#pragma unroll
  for (int sub = 0; sub < 2; ++sub) {
    const int col0 = wave * 32 + sub * 16;
    const int gcol = col0 + n;
    const float* wl_row = W2l + (size_t)gcol * H_DIM;
    const float* wr_row = W2r + (size_t)gcol * H_DIM;

    v8f acc = {};
#pragma unroll 8
    for (int k0 = 0; k0 < H_DIM; k0 += 4) {
      v2f a, b;
      a.x = sM[m][k0 + khalf] * rcpc;
      a.y = sM[m][k0 + khalf + 1] * rcpc;
      b.x = wl_row[k0 + khalf];
      b.y = wl_row[k0 + khalf + 1];
      acc = wmma_f32_k4(a, b, acc);
    }
#pragma unroll 8
    for (int k0 = 0; k0 < H_DIM; k0 += 4) {
      v2f a, b;
      a.x = sH[m][k0 + khalf];
      a.y = sH[m][k0 + khalf + 1];
      b.x = wr_row[k0 + khalf];
      b.y = wr_row[k0 + khalf + 1];
      acc = wmma_f32_k4(a, b, acc);
    }

    const float bias = b2l[gcol];
    float s = 0.0f, sq = 0.0f;
    const int rbase = tile * 16 + ((lane >> 4) << 3);
    if (full) {
#pragma unroll
      for (int r = 0; r < 8; ++r) {
        float v = acc[r] + bias;
        z2[(size_t)(rbase + r) * H_DIM + gcol] = v;
        s += v;
        sq += v * v;
      }
    } else {
      for (int r = 0; r < 8; ++r) {
        int row = rbase + r;
        if (row < N) {
          float v = acc[r] + bias;
          z2[(size_t)row * H_DIM + gcol] = v;
          s += v;
          sq += v * v;
        }
      }
    }
    atomicAdd(&sum2[gcol], s);
    atomicAdd(&sq2[gcol], sq);
  }
}

// ---------------------------------------------------------------------------
// 6) per-channel BN2 scale/shift prep
// ---------------------------------------------------------------------------
__global__ void k_bn_prep(const float* __restrict__ sum,
                          const float* __restrict__ sq,
                          const float* __restrict__ gamma,
                          const float* __restrict__ beta,
                          float* __restrict__ scale,
                          float* __restrict__ shift,
                          float invN) {
  int j = threadIdx.x;
  float mu = sum[j] * invN;
  float var = sq[j] * invN - mu * mu;
  float sc = gamma[j] * rsqrtf(var + EPS);
  scale[j] = sc;
  shift[j] = beta[j] - mu * sc;
}

// ---------------------------------------------------------------------------
// 7) fused BN2 + ReLU + output Linear(256 -> 4): one thread per node
// ---------------------------------------------------------------------------
__global__ void k_output(const float* __restrict__ z2,
                         const float* __restrict__ scale,
                         const float* __restrict__ shift,
                         const float* __restrict__ Wo,
                         const float* __restrict__ bo,
                         float* __restrict__ out,
                         int N) {
  int i = blockIdx.x * blockDim.x + threadIdx.x;
  if (i >= N) return;
  float acc0 = bo[0], acc1 = bo[1], acc2 = bo[2], acc3 = bo[3];
  const float* zr = z2 + (size_t)i * H_DIM;
#pragma unroll 4
  for (int k = 0; k < H_DIM; ++k) {
    float v = fmaxf(zr[k] * scale[k] + shift[k], 0.0f);
    acc0 += v * Wo[0 * H_DIM + k];
    acc1 += v * Wo[1 * H_DIM + k];
    acc2 += v * Wo[2 * H_DIM + k];
    acc3 += v * Wo[3 * H_DIM + k];
  }
  out[(size_t)i * 4 + 0] = acc0;
  out[(size_t)i * 4 + 1] = acc1;
  out[(size_t)i * 4 + 2] = acc2;
  out[(size_t)i * 4 + 3] = acc3;
}

// ---------------------------------------------------------------------------
extern "C" void kernel_launch(void* const* d_in, const int* in_sizes, int n_in,
                              void* d_out, int out_size, void* d_ws, size_t ws_size,
                              hipStream_t stream) {
  const float* x   = (const float*)d_in[0];
  const int*   ei  = (const int*)d_in[1];
  const float* W1l = (const float*)d_in[2];
  const float* b1l = (const float*)d_in[3];
  const float* W1r = (const float*)d_in[4];
  const float* g1  = (const float*)d_in[5];
  const float* be1 = (const float*)d_in[6];
  const float* W2l = (const float*)d_in[7];
  const float* b2l = (const float*)d_in[8];
  const float* W2r = (const float*)d_in[9];
  const float* g2  = (const float*)d_in[10];
  const float* be2 = (const float*)d_in[11];
  const float* Wo  = (const float*)d_in[12];
  const float* bo  = (const float*)d_in[13];
  float* out = (float*)d_out;

  const long N = in_sizes[0] / 4;
  const long E = in_sizes[1] / 2;
  const long NH = N * H_DIM;

  // workspace layout (floats)
  float* ws   = (float*)d_ws;
  float* zbuf = ws;                 // N*H   (z1, later reused as z2)
  float* hbuf = zbuf + NH;          // N*H   (h1 = relu(bn(z1)))
  float* agg2 = hbuf + NH;          // N*H   (layer-2 accumulator; zeroed)
  float* agg1 = agg2 + NH;          // N*4   (layer-1 accumulator; zeroed)
  float* cnt  = agg1 + N * 4;       // N     (in-degree counts; zeroed)
  float* st   = cnt + N;            // 6*H stats: sum1,sq1,sum2,sq2,scale2,shift2
  float* sum1 = st;
  float* sq1  = st + H_DIM;
  float* sum2 = st + 2 * H_DIM;
  float* sq2  = st + 3 * H_DIM;
  float* sc2  = st + 4 * H_DIM;
  float* sh2  = st + 5 * H_DIM;

  const float invN = 1.0f / (float)N;
  const int mtiles = (int)((N + 15) / 16);

  // 0) zero the accumulators + stats (contiguous region: agg2..end of sq2)
  long zn = NH + N * 4 + N + 4 * H_DIM;
  k_zero<<<2048, 256, 0, stream>>>(agg2, zn);

  // 1) layer-1 edge scatter
  k_scatter1<<<(int)((E + 255) / 256), 256, 0, stream>>>(x, ei, agg1, cnt, E);

  // 2) layer-1 WMMA GEMM + BN1 stats -> zbuf (=z1)
  k_layer1_wmma<<<mtiles, 256, 0, stream>>>(x, agg1, cnt, W1l, b1l, W1r,
                                            zbuf, sum1, sq1, (int)N);

  // 3) BN1 + ReLU -> hbuf (=h1)
  k_bn_relu<<<2048, 256, 0, stream>>>(zbuf, sum1, sq1, g1, be1, hbuf, NH, invN);

  // 4) layer-2 edge scatter (heavy: ~1.6 GB, L2-resident)
  k_scatter2<<<(int)((E + 3) / 4), 256, 0, stream>>>(hbuf, ei, agg2, E);

  // 5) layer-2 dual WMMA GEMM (async-LDS staged) + BN2 stats -> zbuf (=z2)
  k_layer2_wmma<<<mtiles, 256, 0, stream>>>(agg2, cnt, hbuf, W2l, b2l, W2r,
                                            zbuf, sum2, sq2, (int)N);

  // 6) BN2 scale/shift
  k_bn_prep<<<1, H_DIM, 0, stream>>>(sum2, sq2, g2, be2, sc2, sh2, invN);

  // 7) fused BN2 + ReLU + output linear
  k_output<<<(int)((N + 255) / 256), 256, 0, stream>>>(zbuf, sc2, sh2, Wo, bo,
                                                       out, (int)N);
}